// YOLOLoss_37855841747437
// MI455X (gfx1250) — compile-verified
//
#include <hip/hip_runtime.h>
#include <hip/hip_bf16.h>

// Problem constants (from reference): B=32, A=3, G=52, C=80, N=512
#define B_      32
#define A_      3
#define G_      52
#define C_      80
#define N_      512
#define GG_     (G_ * G_)
#define NELEM_  (B_ * A_ * GG_)      // 259,584 = 4056 tiles of 64 = 1014 chunks of 256
#define NTILES_ (NELEM_ / 64)
#define NCHUNK_ (NTILES_ / 4)        // 1014, exact
#define NB1     128                  // blocks for the dense reduction
#define T1      256                  // threads per block (8 wave32 waves)

typedef float v2f __attribute__((ext_vector_type(2)));
typedef float v8f __attribute__((ext_vector_type(8)));

// bce helpers:  bce(p,1) = -clip(ln p, -100) ;  bce(p,0) = -clip(ln(1-p), -100)
__device__ __forceinline__ float neg_ln_clip(float x) {
    return -fmaxf(logf(x), -100.0f);
}

// ---------------------------------------------------------------------------
// Kernel 1: dense sum of -clip(ln(1-conf)) over all B*A*G*G cells.
// V_WMMA_F32_16X16X4_F32 with B = ones: D[m,n] = rowsum(A[m,:]) + C[m,n],
// i.e. a 64-wide full-precision f32 accumulator per WMMA issue.
//  - wave index goes through readfirstlane so the trip count is scalar
//    (s_cbranch loop, EXEC stays all-1s for WMMA, no v_cmpx churn)
//  - lane l loads the contiguous pair p[2l],p[2l+1] as one b64 load
//    (slot assignment is irrelevant for a row-sum)
//  - 4 tiles/iter into 4 independent accumulators for ILP
// Deterministic: fixed order per wave + LDS tree reduction (no float atomics).
// ---------------------------------------------------------------------------
__global__ void __launch_bounds__(T1)
conf_sum_kernel(const float* __restrict__ conf, float* __restrict__ partials)
{
    const int lane  = threadIdx.x & 31;
    // wave-uniform wave id -> scalar loop control around the WMMAs
    const int gwave = __builtin_amdgcn_readfirstlane(
                          (int)(blockIdx.x * (T1 / 32) + (threadIdx.x >> 5)));
    const int nwave = NB1 * (T1 / 32);

    float s;
#if __has_builtin(__builtin_amdgcn_wmma_f32_16x16x4_f32)
    v8f acc0 = {0.f,0.f,0.f,0.f,0.f,0.f,0.f,0.f};
    v8f acc1 = acc0, acc2 = acc0, acc3 = acc0;
    const v2f ones = {1.0f, 1.0f};
    for (int t = gwave; t < NCHUNK_; t += nwave) {
        const float* p = conf + (t << 8) + (lane << 1);   // 256 floats / chunk
        v2f r0 = *(const v2f*)(p);          // one global_load_b64 per tile
        v2f r1 = *(const v2f*)(p +  64);
        v2f r2 = *(const v2f*)(p + 128);
        v2f r3 = *(const v2f*)(p + 192);
        v2f a0, a1, a2, a3;
        a0.x = neg_ln_clip(1.0f - r0.x);  a0.y = neg_ln_clip(1.0f - r0.y);
        a1.x = neg_ln_clip(1.0f - r1.x);  a1.y = neg_ln_clip(1.0f - r1.y);
        a2.x = neg_ln_clip(1.0f - r2.x);  a2.y = neg_ln_clip(1.0f - r2.y);
        a3.x = neg_ln_clip(1.0f - r3.x);  a3.y = neg_ln_clip(1.0f - r3.y);
        // 8 args: (neg_a, A, neg_b, B, c_mod, C, reuse_a, reuse_b)
        acc0 = __builtin_amdgcn_wmma_f32_16x16x4_f32(false, a0, false, ones, (short)0, acc0, false, false);
        acc1 = __builtin_amdgcn_wmma_f32_16x16x4_f32(false, a1, false, ones, (short)0, acc1, false, false);
        acc2 = __builtin_amdgcn_wmma_f32_16x16x4_f32(false, a2, false, ones, (short)0, acc2, false, false);
        acc3 = __builtin_amdgcn_wmma_f32_16x16x4_f32(false, a3, false, ones, (short)0, acc3, false, false);
    }
    float t0 = 0.f;
    #pragma unroll
    for (int i = 0; i < 8; ++i)
        t0 += acc0[i] + acc1[i] + acc2[i] + acc3[i];
    // D has 16 identical columns -> sum over all 256 lane-slots = 16 * true sum
    s = t0 * (1.0f / 16.0f);
#else
    s = 0.0f;
    for (int t = gwave; t < NCHUNK_; t += nwave) {
        const float* p = conf + (t << 8) + (lane << 1);
        #pragma unroll
        for (int k = 0; k < 4; ++k) {
            v2f r = *(const v2f*)(p + 64 * k);
            s += neg_ln_clip(1.0f - r.x) + neg_ln_clip(1.0f - r.y);
        }
    }
#endif

    __shared__ float red[T1];
    red[threadIdx.x] = s;
    __syncthreads();
    for (int off = T1 / 2; off > 0; off >>= 1) {
        if (threadIdx.x < off) red[threadIdx.x] += red[threadIdx.x + off];
        __syncthreads();
    }
    if (threadIdx.x == 0) partials[blockIdx.x] = red[0];
}

// ---------------------------------------------------------------------------
// Kernel 2: single block, 512 threads (one thread per target).
// IoU/argmax, duplicate-index dedup (JAX scatter: set = last wins,
// multiply/max = union), sparse gathers, deterministic reduction, final loss.
// ---------------------------------------------------------------------------
__device__ __forceinline__ float block_reduce_f(float v, float* red)
{
    const int t = threadIdx.x;
    red[t] = v;
    __syncthreads();
    for (int off = N_ / 2; off > 0; off >>= 1) {
        if (t < off) red[t] += red[t + off];
        __syncthreads();
    }
    float r = red[0];
    __syncthreads();
    return r;
}

__global__ void __launch_bounds__(N_)
yolo_finish_kernel(const float* __restrict__ pred_cls,
                   const float* __restrict__ cx,
                   const float* __restrict__ cy,
                   const float* __restrict__ w,
                   const float* __restrict__ h,
                   const float* __restrict__ conf,
                   const float* __restrict__ anchors,
                   const float* __restrict__ target,
                   const float* __restrict__ partials, int nPartials,
                   float* __restrict__ out)
{
    __shared__ int   s_cid[N_];     // obj cell id (b,best,gj,gi) flattened
    __shared__ int   s_base[N_];    // cell id with a=0
    __shared__ int   s_lab[N_];
    __shared__ int   s_excl[N_];    // bit a: cell (b,a,gj,gi) excluded from noobj
    __shared__ float s_tx[N_], s_ty[N_], s_tw[N_], s_th[N_];
    __shared__ float red[N_];

    const int n = threadIdx.x;

    // -------- per-target setup --------
    float awv[3] = {anchors[0], anchors[2], anchors[4]};
    float ahv[3] = {anchors[1], anchors[3], anchors[5]};

    const float* tr = target + n * 6;
    const int   lab = (int)tr[1];
    const float gx = tr[2] * (float)G_;
    const float gy = tr[3] * (float)G_;
    const float gw = tr[4] * (float)G_;
    const float gh = tr[5] * (float)G_;
    const int si = (int)tr[0];
    const int gi = (int)floorf(gx);
    const int gj = (int)floorf(gy);

    float iou[3];
    #pragma unroll
    for (int a = 0; a < 3; ++a) {
        float inter = fminf(awv[a], gw) * fminf(ahv[a], gh);
        float uni   = awv[a] * ahv[a] + 1e-16f + gw * gh - inter;
        iou[a] = inter / uni;
    }
    int best = 0;                       // argmax, first max wins (strict >)
    if (iou[1] > iou[0])    best = 1;
    if (iou[2] > iou[best]) best = 2;

    const int base = si * (A_ * GG_) + gj * G_ + gi;
    int em = 0;
    #pragma unroll
    for (int a = 0; a < 3; ++a)
        if (a == best || iou[a] > 0.5f) em |= (1 << a);

    s_cid[n]  = base + best * GG_;
    s_base[n] = base;
    s_lab[n]  = lab;
    s_excl[n] = em;
    s_tx[n] = gx - floorf(gx);
    s_ty[n] = gy - floorf(gy);
    s_tw[n] = logf(gw / awv[best] + 1e-16f);
    s_th[n] = logf(gh / ahv[best] + 1e-16f);
    __syncthreads();

    // -------- duplicate handling over obj cells --------
    const int mycid = s_cid[n];
    bool lastCell  = true;  // no later target hits my cell  (scatter-set winner)
    bool firstCell = true;  // no earlier target hits my cell (count cell once)
    bool lastPair  = true;  // no later target with same (cell,label)
    for (int m = 0; m < N_; ++m) {
        if (s_cid[m] == mycid) {
            if (m > n) {
                lastCell = false;
                if (s_lab[m] == lab) lastPair = false;
            }
            if (m < n) firstCell = false;
        }
    }

    float ax = 0.f, ay = 0.f, aww = 0.f, ahh = 0.f, aco = 0.f, acls = 0.f;
    float nobj = 0.f;

    if (lastCell) {                      // winner supplies the regression target
        float dx = cx[mycid] - s_tx[n];  ax  = dx * dx;
        float dy = cy[mycid] - s_ty[n];  ay  = dy * dy;
        float dw = w[mycid]  - s_tw[n];  aww = dw * dw;
        float dh = h[mycid]  - s_th[n];  ahh = dh * dh;
    }
    if (firstCell) {                     // once per unique obj cell
        nobj = 1.f;
        aco  = neg_ln_clip(conf[mycid]);                 // bce(conf, 1)
        const float* pc = pred_cls + (long)mycid * C_;
        float sc = 0.f;
        for (int c = 0; c < C_; ++c)                     // tcls=0 baseline
            sc += neg_ln_clip(1.0f - pc[c]);
        acls = sc;
    }
    if (lastPair) {                      // once per unique (cell,label): tcls=1
        float p = pred_cls[(long)mycid * C_ + lab];
        acls += neg_ln_clip(p) - neg_ln_clip(1.0f - p);
    }

    // -------- noobj exclusion set (union over targets/anchors) --------
    float aex = 0.f, nex = 0.f;
    for (int k = n; k < 3 * N_; k += N_) {
        const int nn = k / 3, a = k - nn * 3;
        if ((s_excl[nn] >> a) & 1) {
            const int e = s_base[nn] + a * GG_;
            bool uniq = true;
            for (int m = 0; m < N_ && uniq; ++m) {
                const int emask = s_excl[m];
                const int be    = s_base[m];
                const int kb    = m * 3;
                #pragma unroll
                for (int a2 = 0; a2 < 3; ++a2) {
                    if (kb + a2 >= k) break;             // only earlier items
                    if (((emask >> a2) & 1) && (be + a2 * GG_) == e) {
                        uniq = false; break;
                    }
                }
            }
            if (uniq) {
                nex += 1.f;
                aex += neg_ln_clip(1.0f - conf[e]);      // bce(conf, 0)
            }
        }
    }

    // -------- deterministic block reductions --------
    const float SX   = block_reduce_f(ax,   red);
    const float SY   = block_reduce_f(ay,   red);
    const float SW   = block_reduce_f(aww,  red);
    const float SH   = block_reduce_f(ahh,  red);
    const float SCO  = block_reduce_f(aco,  red);
    const float SCLS = block_reduce_f(acls, red);
    const float SEX  = block_reduce_f(aex,  red);
    const float NOBJ = block_reduce_f(nobj, red);
    const float NEX  = block_reduce_f(nex,  red);

    if (threadIdx.x == 0) {
        float total = 0.f;                               // dense Σ -ln(1-conf)
        for (int i = 0; i < nPartials; ++i) total += partials[i];

        const float n_noobj   = (float)NELEM_ - NEX;
        const float sum_noobj = total - SEX;

        const float loss =
            (SX + SY + SW + SH) / NOBJ                       // x+y+w+h mse
          + 1.0f   * (SCO / NOBJ)                            // OBJ_SCALE * conf_obj
          + 100.0f * (sum_noobj / n_noobj)                   // NOOBJ_SCALE * conf_noobj
          + SCLS / (NOBJ * (float)C_);                       // cls
        out[0] = loss;
    }
}

// ---------------------------------------------------------------------------
extern "C" void kernel_launch(void* const* d_in, const int* in_sizes, int n_in,
                              void* d_out, int out_size, void* d_ws, size_t ws_size,
                              hipStream_t stream)
{
    (void)in_sizes; (void)n_in; (void)out_size; (void)ws_size;
    // setup_inputs order:
    // 0: pred_boxes (unused), 1: pred_cls, 2: center_x, 3: center_y,
    // 4: width, 5: height, 6: confidence, 7: anchors, 8: target
    const float* pred_cls = (const float*)d_in[1];
    const float* cx       = (const float*)d_in[2];
    const float* cy       = (const float*)d_in[3];
    const float* w        = (const float*)d_in[4];
    const float* h        = (const float*)d_in[5];
    const float* conf     = (const float*)d_in[6];
    const float* anchors  = (const float*)d_in[7];
    const float* target   = (const float*)d_in[8];

    float* partials = (float*)d_ws;          // NB1 floats of scratch

    conf_sum_kernel<<<NB1, T1, 0, stream>>>(conf, partials);
    yolo_finish_kernel<<<1, N_, 0, stream>>>(pred_cls, cx, cy, w, h, conf,
                                             anchors, target,
                                             partials, NB1, (float*)d_out);
}